// gcn_module_38500086841953
// MI455X (gfx1250) — compile-verified
//
#include <hip/hip_runtime.h>

// GCN forward on MI455X (gfx1250), fp32 end-to-end via V_WMMA_F32_16X16X4_F32.
//
//   support = x @ W            (2.1 GFLOP, staged into d_ws: 64 MiB fp32)
//   out     = adj @ support    (68.7 GFLOP, dominant -> tiled WMMA GEMM)
//
// AI ~260 FLOP/byte vs 23.3 TB/s HBM -> compute-bound on the f32 WMMA pipe.
// Kernel 2 uses gfx1250 GLOBAL_LOAD_ASYNC_TO_LDS_B128 (ASYNCcnt) to stage
// adj/support chunks into double-buffered LDS with zero VGPR traffic, fully
// overlapped with the 64-WMMA compute of the previous chunk.

typedef __attribute__((ext_vector_type(2))) float v2f;
typedef __attribute__((ext_vector_type(4))) float f4;
typedef __attribute__((ext_vector_type(8))) float v8f;
typedef __attribute__((ext_vector_type(4))) int   v4i;

constexpr int BATCH = 8;
constexpr int D1    = 16;
constexpr int NN    = 2048;
constexpr int F     = 64;      // F_IN == F_OUT == 64
constexpr int MBLK  = 128;     // rows per workgroup (8 waves x 16 rows)
constexpr int KC    = 64;      // K chunk staged in LDS
constexpr int LSA   = KC + 4;  // A stride: 68 == 4 (mod 64) -> b64 frag loads hit 64 distinct banks
constexpr int LSB   = F + 8;   // B stride: 72 == 8 (mod 32) -> lane halves on disjoint bank groups

#if __has_builtin(__builtin_amdgcn_global_load_async_to_lds_b128)
#define HAVE_ASYNC_LDS 1
#else
#define HAVE_ASYNC_LDS 0
#endif

// Copy 16 bytes global -> LDS. Async (data-mover, ASYNCcnt) when available,
// else synchronous VGPR round-trip fallback.
__device__ __forceinline__ void copy_b128(const float* g, float* l) {
#if HAVE_ASYNC_LDS
  __builtin_amdgcn_global_load_async_to_lds_b128(
      (__attribute__((address_space(1))) v4i*)g,
      (__attribute__((address_space(3))) v4i*)l,
      /*imm_offset=*/0, /*cpol=*/0);
#else
  *(f4*)l = *(const f4*)g;
#endif
}

__device__ __forceinline__ void wait_async_done() {
#if HAVE_ASYNC_LDS
#if __has_builtin(__builtin_amdgcn_s_wait_asynccnt)
  __builtin_amdgcn_s_wait_asynccnt(0);
#else
  asm volatile("s_wait_asynccnt 0" ::: "memory");
#endif
#endif
}

// ---------------------------------------------------------------------------
// Kernel 1: support[r, g] = sum_f x[r, f] * W[f, g]   for r in [0, B*D1*N)
// One wave computes a 16x64 output strip (K = 64 -> 16 WMMA steps x 4 N-tiles).
// ---------------------------------------------------------------------------
__global__ __launch_bounds__(256) void gcn_transform(const float* __restrict__ x,
                                                     const float* __restrict__ w,
                                                     float* __restrict__ supp) {
  __shared__ float Wt[F * LSA];          // W transposed: Wt[g*LSA + f] = W[f][g]
  const int t = threadIdx.x;
  for (int idx = t; idx < F * F; idx += 256) {
    const int f = idx >> 6, g = idx & 63;
    Wt[g * LSA + f] = w[idx];
  }
  __syncthreads();

  const int lane = t & 31, wv = t >> 5;
  const int rr = lane & 15;              // row within 16-row tile (ISA A/B lane map)
  const int hh = lane >> 4;              // lane half: selects K pair / M+8 rows
  const int r0 = blockIdx.x * MBLK + wv * 16;

  const float* xrow = x + (size_t)(r0 + rr) * F + 2 * hh;   // A frag: row rr, K pair {0,1}/{2,3}
  v8f c[4] = {};
#pragma unroll
  for (int k = 0; k < F; k += 4) {
    const v2f a = *(const v2f*)(xrow + k);
#pragma unroll
    for (int g = 0; g < 4; ++g) {
      // B frag VGPR v holds row k+v+2*hh, col 16g+rr -> contiguous b64 in transposed LDS
      const v2f bfrag = *(const v2f*)(&Wt[(16 * g + rr) * LSA + k + 2 * hh]);
      c[g] = __builtin_amdgcn_wmma_f32_16x16x4_f32(false, a, false, bfrag,
                                                   (short)0, c[g], false, false);
    }
  }

  // C/D layout: VGPR v -> row v (lanes 0-15) / row v+8 (lanes 16-31), col = lane%16
  float* orow = supp + (size_t)(r0 + 8 * hh) * F + rr;
#pragma unroll
  for (int g = 0; g < 4; ++g)
#pragma unroll
    for (int v = 0; v < 8; ++v)
      orow[(size_t)v * F + 16 * g] = c[g][v];
}

// ---------------------------------------------------------------------------
// Kernel 2: out[b,i, m0:m0+128, :] = adj[b, m0:m0+128, :] @ support[b,i]
// 8 waves x (16x64 strip). K in 64-chunks, double-buffered async LDS staging:
//   wait_asynccnt 0 ; barrier ; issue async loads chunk p+1 ; 64 WMMAs chunk p
// ---------------------------------------------------------------------------
__global__ __launch_bounds__(256) void gcn_aggregate(const float* __restrict__ adj,
                                                     const float* __restrict__ supp,
                                                     float* __restrict__ out) {
  __shared__ float As[2][MBLK * LSA];    // adj chunk, [m][k], stride 68   (2x34 KB)
  __shared__ float Bs[2][KC * LSB];      // support chunk, [k][g], stride 72 (2x18 KB)

  const int m0 = blockIdx.x * MBLK;
  const int i  = blockIdx.y;
  const int b  = blockIdx.z;
  const float* adj_b = adj + (size_t)b * NN * NN;
  const float* sp_bi = supp + (size_t)(b * D1 + i) * NN * F;

  const int t = threadIdx.x, lane = t & 31, wv = t >> 5;
  const int rr = lane & 15, hh = lane >> 4;

  auto stage = [&](int buf, int k0) {
    // adj chunk: 128x64 floats = 2048 b128 transfers (8 per thread), coalesced
    for (int idx = t; idx < MBLK * (KC / 4); idx += 256) {
      const int row = idx >> 4, col4 = (idx & 15) * 4;
      copy_b128(adj_b + (size_t)(m0 + row) * NN + k0 + col4,
                &As[buf][row * LSA + col4]);
    }
    // support chunk: 64x64 floats = 1024 b128 transfers (4 per thread)
    for (int idx = t; idx < KC * (F / 4); idx += 256) {
      const int row = idx >> 4, col4 = (idx & 15) * 4;
      copy_b128(sp_bi + (size_t)(k0 + row) * F + col4,
                &Bs[buf][row * LSB + col4]);
    }
  };

  v8f c[4] = {};
  int cur = 0;
  stage(0, 0);

  for (int k0 = 0; k0 < NN; k0 += KC) {
    wait_async_done();    // chunk p resident in As/Bs[cur]
    __syncthreads();      // all waves past their waits; buf cur^1 consumers done

    if (k0 + KC < NN) stage(cur ^ 1, k0 + KC);   // overlap with WMMAs below

    const float* arow = &As[cur][(wv * 16 + rr) * LSA + 2 * hh];
    const float* bbase = &Bs[cur][2 * hh * LSB + rr];
#pragma unroll
    for (int kk = 0; kk < KC; kk += 4) {
      const v2f a = *(const v2f*)(arow + kk);    // ds_load_b64, 64 distinct banks
#pragma unroll
      for (int g = 0; g < 4; ++g) {
        // B frag VGPR v holds row kk+v+2*hh, col 16g+rr (un-transposed LDS)
        v2f bfrag;
        bfrag.x = bbase[(kk + 0) * LSB + 16 * g];
        bfrag.y = bbase[(kk + 1) * LSB + 16 * g];
        c[g] = __builtin_amdgcn_wmma_f32_16x16x4_f32(false, a, false, bfrag,
                                                     (short)0, c[g], false, false);
      }
    }
    cur ^= 1;
  }

  float* orow = out + ((size_t)(b * D1 + i) * NN + m0 + wv * 16 + 8 * hh) * F + rr;
#pragma unroll
  for (int g = 0; g < 4; ++g)
#pragma unroll
    for (int v = 0; v < 8; ++v)
      orow[(size_t)v * F + 16 * g] = c[g][v];
}

// ---------------------------------------------------------------------------
extern "C" void kernel_launch(void* const* d_in, const int* in_sizes, int n_in,
                              void* d_out, int out_size, void* d_ws, size_t ws_size,
                              hipStream_t stream) {
  const float* x   = (const float*)d_in[0];   // [8,16,2048,64]
  const float* adj = (const float*)d_in[1];   // [8,2048,2048]
  const float* w   = (const float*)d_in[2];   // [64,64]
  float* out  = (float*)d_out;                // [8,16,2048,64]
  float* supp = (float*)d_ws;                 // needs 8*16*2048*64*4 = 64 MiB scratch

  (void)in_sizes; (void)n_in; (void)out_size; (void)ws_size;

  dim3 g1((BATCH * D1 * NN) / MBLK);          // 2048 blocks
  gcn_transform<<<g1, 256, 0, stream>>>(x, w, supp);

  dim3 g2(NN / MBLK, D1, BATCH);              // (16, 16, 8)
  gcn_aggregate<<<g2, 256, 0, stream>>>(adj, supp, out);
}